// PwGaANLayer_8169027797701
// MI455X (gfx1250) — compile-verified
//
#include <hip/hip_runtime.h>
#include <math.h>

// PwGaAN layer for MI455X (gfx1250, wave32).
//  - z = v@Wa and zg = v@Wg via V_WMMA_F32_16X16X4_F32 (exact fp32 matrix path)
//  - attention logits factorized to per-node scalars s1 = z.a1, s2 = z.a2
//  - edge phase: HW float atomics (add) + int-trick float atomic max
// Workspace layout (floats), total = 64N*5 + 5N + E ~ 17.05M floats ~ 68.2 MB.

typedef __attribute__((ext_vector_type(2))) float v2f;
typedef __attribute__((ext_vector_type(8))) float v8f;

#define DIM 64

__device__ __forceinline__ void atomicAddF(float* addr, float val) {
    __hip_atomic_fetch_add(addr, val, __ATOMIC_RELAXED, __HIP_MEMORY_SCOPE_AGENT);
}

// IEEE-ordered float atomic max via integer atomics (maps to HW int atomics).
// Target must be initialized to -inf.
__device__ __forceinline__ void atomicMaxF(float* addr, float val) {
    if (val >= 0.0f) {
        atomicMax((int*)addr, __float_as_int(val));
    } else {
        atomicMin((unsigned int*)addr, __float_as_uint(val));
    }
}

__global__ void fill_kernel(float* __restrict__ p, float val, size_t n) {
    size_t i = (size_t)blockIdx.x * blockDim.x + threadIdx.x;
    size_t stride = (size_t)gridDim.x * blockDim.x;
    for (; i < n; i += stride) p[i] = val;
}

// One wave computes a 16x64 tile of BOTH z = v@Wa and zg = v@Wg.
// A (16x4 f32): lane = {m = lane&15}, K = 4*step + 2*(lane>>4) + {0,1} (2 VGPRs)
// B (4x16 f32): lane = {n = lane&15}, K = 4*step + 2*(lane>>4) + {0,1}
// C/D (16x16 f32): vgpr r holds row m = r + 8*(lane>>4), col n = lane&15
__global__ void gemm_wmma_kernel(const float* __restrict__ v,
                                 const float* __restrict__ Wa,
                                 const float* __restrict__ Wg,
                                 float* __restrict__ z,
                                 float* __restrict__ zg,
                                 int N, int ntiles) {
    int wave = blockIdx.x * (blockDim.x >> 5) + (threadIdx.x >> 5);
    if (wave >= ntiles) return;
    int lane = threadIdx.x & 31;
    int n16  = lane & 15;
    int half = lane >> 4;
    int node0 = wave * 16;

    int arow = node0 + n16;
    if (arow >= N) arow = N - 1;               // clamp (benign for ragged tail)
    const float* __restrict__ A = v + (size_t)arow * DIM;

    v8f accA[4], accG[4];
#pragma unroll
    for (int j = 0; j < 4; ++j)
#pragma unroll
        for (int r = 0; r < 8; ++r) { accA[j][r] = 0.0f; accG[j][r] = 0.0f; }

#pragma unroll
    for (int k = 0; k < 16; ++k) {
        int kb = k * 4 + half * 2;
        v2f a;
        a.x = A[kb];
        a.y = A[kb + 1];
#pragma unroll
        for (int j = 0; j < 4; ++j) {
            int col = j * 16 + n16;
            v2f bA, bG;
            bA.x = Wa[(size_t)kb * DIM + col];
            bA.y = Wa[(size_t)(kb + 1) * DIM + col];
            bG.x = Wg[(size_t)kb * DIM + col];
            bG.y = Wg[(size_t)(kb + 1) * DIM + col];
            accA[j] = __builtin_amdgcn_wmma_f32_16x16x4_f32(
                false, a, false, bA, (short)0, accA[j], false, false);
            accG[j] = __builtin_amdgcn_wmma_f32_16x16x4_f32(
                false, a, false, bG, (short)0, accG[j], false, false);
        }
    }

#pragma unroll
    for (int j = 0; j < 4; ++j)
#pragma unroll
        for (int r = 0; r < 8; ++r) {
            int row = node0 + r + 8 * half;
            if (row < N) {
                size_t idx = (size_t)row * DIM + j * 16 + n16;
                z[idx]  = accA[j][r];
                zg[idx] = accG[j][r];
            }
        }
}

// s1[n] = z[n] . att_w[0:64], s2[n] = z[n] . att_w[64:128]
__global__ void node_scores_kernel(const float* __restrict__ z,
                                   const float* __restrict__ att_w,
                                   float* __restrict__ s1,
                                   float* __restrict__ s2, int N) {
    __shared__ float r1[DIM];
    __shared__ float r2[DIM];
    int n = blockIdx.x;
    int t = threadIdx.x;
    float zv = z[(size_t)n * DIM + t];
    r1[t] = zv * att_w[t];
    r2[t] = zv * att_w[DIM + t];
    __syncthreads();
#pragma unroll
    for (int s = 32; s > 0; s >>= 1) {
        if (t < s) { r1[t] += r1[t + s]; r2[t] += r2[t + s]; }
        __syncthreads();
    }
    if (t == 0) { s1[n] = r1[0]; s2[n] = r2[0]; }
}

// e = leaky_relu(pre_w*s1[src] + s2[dst]); segment max + degree
__global__ void edge_pass1(const float* __restrict__ pre_w,
                           const int* __restrict__ src,
                           const int* __restrict__ dst,
                           const float* __restrict__ s1,
                           const float* __restrict__ s2,
                           float* __restrict__ e_buf,
                           float* __restrict__ e_max,
                           unsigned* __restrict__ deg, int E) {
    int e = blockIdx.x * blockDim.x + threadIdx.x;
    if (e >= E) return;
    int s = src[e], t = dst[e];
    float ee = pre_w[e] * s1[s] + s2[t];
    ee = (ee > 0.0f) ? ee : 0.01f * ee;        // leaky_relu, slope 0.01
    e_buf[e] = ee;
    atomicMaxF(&e_max[t], ee);
    atomicAdd(&deg[t], 1u);
}

// ex = exp(e - e_max[dst]); segment sum denom; e_buf <- ex
__global__ void edge_pass2(const int* __restrict__ dst,
                           float* __restrict__ e_buf,
                           const float* __restrict__ e_max,
                           float* __restrict__ denom, int E) {
    int e = blockIdx.x * blockDim.x + threadIdx.x;
    if (e >= E) return;
    int t = dst[e];
    float ex = expf(e_buf[e] - e_max[t]);
    e_buf[e] = ex;
    atomicAddF(&denom[t], ex);
}

// 64 threads per edge: vector scatter of h, sumpw, maxf
__global__ void edge_pass3(const float* __restrict__ v,
                           const float* __restrict__ z,
                           const float* __restrict__ zg,
                           const float* __restrict__ pre_w,
                           const int* __restrict__ src,
                           const int* __restrict__ dst,
                           const float* __restrict__ e_buf,
                           const float* __restrict__ denom,
                           float* __restrict__ h,
                           float* __restrict__ maxf,
                           float* __restrict__ sumpw, int E) {
    int e = blockIdx.x * 4 + (threadIdx.x >> 6);
    if (e >= E) return;
    int d = threadIdx.x & 63;
    int s = src[e], t = dst[e];
    float alpha = e_buf[e] / fmaxf(denom[t], 1e-30f);
    float pw = pre_w[e];
    size_t si = (size_t)s * DIM + d;
    size_t ti = (size_t)t * DIM + d;
    atomicAddF(&h[ti],     alpha * z[si]);
    atomicAddF(&sumpw[ti], pw * v[si]);
    atomicMaxF(&maxf[ti],  pw * zg[si]);
}

// gate = sigmoid([v, max_feat, mean_feat] . gate_w); out = gate * h
__global__ void finalize_kernel(const float* __restrict__ v,
                                const float* __restrict__ gate_w,
                                const float* __restrict__ h,
                                const float* __restrict__ maxf,
                                const float* __restrict__ sumpw,
                                const unsigned* __restrict__ deg,
                                float* __restrict__ out, int N) {
    __shared__ float red[DIM];
    int n = blockIdx.x;
    int d = threadIdx.x;
    size_t idx = (size_t)n * DIM + d;
    float degn = (float)deg[n];
    float mf = (degn > 0.0f) ? maxf[idx] : 0.0f;          // where(deg>0, max, 0)
    float mean = sumpw[idx] / fmaxf(degn, 1.0f);           // / max(deg, 1)
    red[d] = v[idx] * gate_w[d] + mf * gate_w[DIM + d] + mean * gate_w[2 * DIM + d];
    __syncthreads();
#pragma unroll
    for (int s = 32; s > 0; s >>= 1) {
        if (d < s) red[d] += red[d + s];
        __syncthreads();
    }
    float g = 1.0f / (1.0f + expf(-red[0]));
    out[idx] = g * h[idx];
}

extern "C" void kernel_launch(void* const* d_in, const int* in_sizes, int n_in,
                              void* d_out, int out_size, void* d_ws, size_t ws_size,
                              hipStream_t stream) {
    const float* v      = (const float*)d_in[0];
    const float* pre_w  = (const float*)d_in[1];
    const float* Wa     = (const float*)d_in[2];
    const float* att_w  = (const float*)d_in[3];
    const float* Wg     = (const float*)d_in[4];
    const float* gate_w = (const float*)d_in[5];
    const int*   src    = (const int*)d_in[6];
    const int*   dst    = (const int*)d_in[7];
    float* out = (float*)d_out;

    const int N = in_sizes[0] / DIM;
    const int E = in_sizes[6];

    // ---- workspace carve (contiguous zero block, then contiguous -inf block)
    float* ws    = (float*)d_ws;
    float* z     = ws;                            // 64N
    float* zg    = z  + (size_t)N * DIM;          // 64N
    float* s1    = zg + (size_t)N * DIM;          // N
    float* s2    = s1 + N;                        // N
    float* denom = s2 + N;                        // N   -- zero block start
    unsigned* deg = (unsigned*)(denom + N);       // N   (0u == 0.0f bits)
    float* h     = (float*)(deg + N);             // 64N
    float* sumpw = h + (size_t)N * DIM;           // 64N -- zero block end (130N)
    float* e_max = sumpw + (size_t)N * DIM;       // N   -- -inf block start
    float* maxf  = e_max + N;                     // 64N -- -inf block end (65N)
    float* e_buf = maxf + (size_t)N * DIM;        // E   (no init needed)
    (void)e_buf; (void)ws_size; (void)n_in; (void)out_size;

    const float NEG_INF = -__builtin_inff();
    fill_kernel<<<2048, 256, 0, stream>>>(denom, 0.0f, (size_t)N * 130);
    fill_kernel<<<2048, 256, 0, stream>>>(e_max, NEG_INF, (size_t)N * 65);

    const int ntiles = (N + 15) / 16;
    const int gemmBlocks = (ntiles + 7) / 8;      // 8 waves (tiles) per 256-thr block
    gemm_wmma_kernel<<<gemmBlocks, 256, 0, stream>>>(v, Wa, Wg, z, zg, N, ntiles);

    node_scores_kernel<<<N, DIM, 0, stream>>>(z, att_w, s1, s2, N);

    edge_pass1<<<(E + 255) / 256, 256, 0, stream>>>(pre_w, src, dst, s1, s2,
                                                    e_buf, e_max, deg, E);
    edge_pass2<<<(E + 255) / 256, 256, 0, stream>>>(dst, e_buf, e_max, denom, E);
    edge_pass3<<<(E + 3) / 4, 256, 0, stream>>>(v, z, zg, pre_w, src, dst,
                                                e_buf, denom, h, maxf, sumpw, E);

    finalize_kernel<<<N, DIM, 0, stream>>>(v, gate_w, h, maxf, sumpw, deg, out, N);
}